// MaceNet_71983651881018
// MI455X (gfx1250) — compile-verified
//
#include <hip/hip_runtime.h>

// ---------------------------------------------------------------------------
// MACE-like message passing net for MI455X (gfx1250), wave32, f32 WMMA.
// One workgroup per batch element; all intermediates LDS-resident.
// ---------------------------------------------------------------------------

typedef __attribute__((ext_vector_type(2))) float v2f;
typedef __attribute__((ext_vector_type(8))) float v8f;

#define WMMA_F32(a, b, c) \
  __builtin_amdgcn_wmma_f32_16x16x4_f32(false, (a), false, (b), (short)0, (c), false, false)

__device__ __forceinline__ v8f zero8() {
  v8f z = {0.f, 0.f, 0.f, 0.f, 0.f, 0.f, 0.f, 0.f};
  return z;
}

__device__ __forceinline__ float siluf(float x) { return x / (1.f + expf(-x)); }
__device__ __forceinline__ float eluf(float x)  { return x > 0.f ? x : expf(x) - 1.f; }

__device__ __forceinline__ void lds_atomic_add(float* p, float v) {
  __hip_atomic_fetch_add(p, v, __ATOMIC_RELAXED, __HIP_MEMORY_SCOPE_WORKGROUP);
}

// C[MxN] = A[MxK] * B[KxN], 16x16 tiles distributed over waves.
// A, C typically LDS; B typically global (L2-resident weights).
__device__ __forceinline__ void gemm_tiles(const float* __restrict__ A, int lda,
                                           const float* __restrict__ B, int ldb,
                                           float* __restrict__ C, int ldc,
                                           int M, int N, int K,
                                           int wave, int nwaves, int ln, int hi) {
  const int tn_cnt = N >> 4;
  const int tiles = (M >> 4) * tn_cnt;
  for (int tile = wave; tile < tiles; tile += nwaves) {
    const int tm = (tile / tn_cnt) << 4;
    const int tn = (tile % tn_cnt) << 4;
    v8f acc = zero8();
    const float* Arow = A + (tm + ln) * lda + hi * 2;
    const float* Bcol = B + (hi * 2) * ldb + tn + ln;
    for (int k = 0; k < K; k += 4) {
      v2f a, b;
      a.x = Arow[k];
      a.y = Arow[k + 1];
      b.x = Bcol[k * ldb];
      b.y = Bcol[k * ldb + ldb];
      acc = WMMA_F32(a, b, acc);
    }
#pragma unroll
    for (int j = 0; j < 8; ++j)
      C[(tm + j + hi * 8) * ldc + tn + ln] = acc[j];
  }
}

__global__ __launch_bounds__(512, 1)
void mace_kernel(const float* __restrict__ x,      // (128,32,3)
                 const float* __restrict__ embed,  // (128)
                 const float* __restrict__ W_r1,   // (2,8,64)
                 const float* __restrict__ W_r2,   // (2,64,384)
                 const float* __restrict__ Ws1,    // (2,128,128)
                 const float* __restrict__ Wvd,    // (2,64,128)
                 const float* __restrict__ Ws2,    // (2,128,64)
                 const float* __restrict__ Wv1,    // (2,64,64)
                 const float* __restrict__ U_s,    // (2,192,128)
                 const float* __restrict__ G_sv,   // (2,128,64)
                 const float* __restrict__ U_v,    // (2,64,64)
                 const float* __restrict__ W_ro_s, // (128,128)
                 const float* __restrict__ W_ro_v, // (64,64)
                 const float* __restrict__ W_fin,  // (128,128)
                 const float* __restrict__ b_fin,  // (128)
                 const int* __restrict__ senders,  // (992)
                 const int* __restrict__ receivers,// (992)
                 float* __restrict__ out_vec,      // (128,32,64,3)
                 float* __restrict__ out_inv)      // (128,32,128)
{
  const int tid  = threadIdx.x;
  const int lane = tid & 31;
  const int ln   = lane & 15;
  const int hi   = lane >> 4;
  const int wave = tid >> 5;
  const int NW   = 16;          // 512 threads = 16 wave32
  const int b    = blockIdx.x;

  __shared__ float xs[96];            // positions (32,3)
  __shared__ float com[3];            // center of mass
  __shared__ float hs[32 * 128];      // scalar node features
  __shared__ float hv[3][32 * 64];    // vector node features, component-major
  __shared__ float As[32 * 128];      // scalar aggregation
  __shared__ float Av[3][32 * 64];    // vector aggregation
  __shared__ float P1[32 * 128];      // h_s @ Ws1
  __shared__ float Hd[3][32 * 128];   // h_v[:,:,c] @ Wvd   (Hd[0] reused as gate later)
  __shared__ float P2[32 * 64];       // h_s @ Ws2          (reused as vv later)
  __shared__ float Qb[3][32 * 64];    // h_v[:,:,c] @ Wv1
  __shared__ float wr1s[8 * 64];      // staged W_r1[t]
  __shared__ float Hbuf[32 * 64];     // silu(rb @ W_r1) for 32 edges
  __shared__ float ey[3][32];         // Y1 per edge
  __shared__ int   esi[32], eri[32];  // sender/receiver per edge
  __shared__ float erb[32 * 8];       // bessel basis per edge

  if (tid == 0) {
    __builtin_prefetch(W_r2, 0, 1);
    __builtin_prefetch(Ws1, 0, 1);
    __builtin_prefetch(U_s, 0, 1);
  }

  // ---- init: positions, h_s = embed, h_v = 0 -------------------------------
  if (tid < 96) xs[tid] = x[b * 96 + tid];
  for (int i = tid; i < 32 * 128; i += 512) hs[i] = embed[i & 127];
  {
    float* hvf = &hv[0][0];
    for (int i = tid; i < 3 * 32 * 64; i += 512) hvf[i] = 0.f;
  }
  __syncthreads();
  if (tid < 3) {
    float s = 0.f;
    for (int n = 0; n < 32; ++n) s += xs[n * 3 + tid];
    com[tid] = s * (1.f / 32.f);
  }

  for (int t = 0; t < 2; ++t) {
    const float* Wr1t = W_r1 + t * 8 * 64;
    const float* Wr2t = W_r2 + t * 64 * 384;
    const float* Ws1t = Ws1 + t * 128 * 128;
    const float* Wvdt = Wvd + t * 64 * 128;
    const float* Ws2t = Ws2 + t * 128 * 64;
    const float* Wv1t = Wv1 + t * 64 * 64;
    const float* Ust  = U_s + t * 192 * 128;
    const float* Gsvt = G_sv + t * 128 * 64;
    const float* Uvt  = U_v + t * 64 * 64;

    // stage W_r1[t], zero aggregation buffers
    wr1s[tid] = Wr1t[tid];  // exactly 512 elements
    for (int i = tid; i < 32 * 128; i += 512) As[i] = 0.f;
    {
      float* avf = &Av[0][0];
      for (int i = tid; i < 3 * 32 * 64; i += 512) avf[i] = 0.f;
    }

    // ---- per-node precompute (gather-then-GEMM == GEMM-then-gather) --------
    gemm_tiles(hs, 128, Ws1t, 128, P1, 128, 32, 128, 128, wave, NW, ln, hi);
    for (int c = 0; c < 3; ++c)
      gemm_tiles(&hv[c][0], 64, Wvdt, 128, &Hd[c][0], 128, 32, 128, 64, wave, NW, ln, hi);
    gemm_tiles(hs, 128, Ws2t, 64, P2, 64, 32, 64, 128, wave, NW, ln, hi);
    for (int c = 0; c < 3; ++c)
      gemm_tiles(&hv[c][0], 64, Wv1t, 64, &Qb[c][0], 64, 32, 64, 64, wave, NW, ln, hi);
    __syncthreads();

    // ---- edge phase: 31 iterations x 32 edges ------------------------------
    for (int it = 0; it < 31; ++it) {
      const int e0 = it * 32;
      if (tid < 32) {  // geometry for 32 edges
        const int e = e0 + tid;
        const int s = senders[e], r = receivers[e];
        esi[tid] = s; eri[tid] = r;
        const float dx = xs[r * 3 + 0] - xs[s * 3 + 0];
        const float dy = xs[r * 3 + 1] - xs[s * 3 + 1];
        const float dz = xs[r * 3 + 2] - xs[s * 3 + 2];
        const float rr = sqrtf(dx * dx + dy * dy + dz * dz);
        const float rc = fmaxf(rr, 1e-9f);
        const float ir = 1.f / rc;
        const float s3 = 1.7320508075688772f;
        ey[0][tid] = s3 * dx * ir;
        ey[1][tid] = s3 * dy * ir;
        ey[2][tid] = s3 * dz * ir;
        const float u  = fminf(fmaxf(rr * 0.1f, 0.f), 1.f - 1e-6f);
        const float u2 = u * u;
        const float env = (rr < 10.f) ? expf(-u2 / (1.f - u2)) : 0.f;
        const float base = 0.44721359549995793f * ir * env;  // sqrt(2/10)/r * env
        for (int k = 0; k < 8; ++k)
          erb[tid * 8 + k] = base * sinf((float)(k + 1) * 0.31415926535897931f * rc);
      }
      __syncthreads();

      // GEMM1: H = silu(rb(32x8) @ W_r1(8x64)); 8 tiles
      for (int tile = wave; tile < 8; tile += NW) {
        const int tm = (tile >> 2) << 4;
        const int tn = (tile & 3) << 4;
        v8f acc = zero8();
        for (int k = 0; k < 8; k += 4) {
          v2f a, bb;
          a.x = erb[(tm + ln) * 8 + k + hi * 2];
          a.y = erb[(tm + ln) * 8 + k + hi * 2 + 1];
          bb.x = wr1s[(k + hi * 2) * 64 + tn + ln];
          bb.y = wr1s[(k + hi * 2 + 1) * 64 + tn + ln];
          acc = WMMA_F32(a, bb, acc);
        }
#pragma unroll
        for (int j = 0; j < 8; ++j)
          Hbuf[(tm + j + hi * 8) * 64 + tn + ln] = siluf(acc[j]);
      }
      __syncthreads();

      // GEMM2 + fused scatter: gates G = H(32x64) @ W_r2(64x384), consumed
      // immediately. 24 jobs: 2 edge row-tiles x (8 f-pairs + 4 v-pairs).
      for (int job = wave; job < 24; job += NW) {
        const int et = job / 12;
        const int p  = job % 12;
        const float* Arow = Hbuf + (et * 16 + ln) * 64 + hi * 2;
        if (p < 8) {  // g_ss (cols p*16..) paired with g_vs (cols 128+p*16..)
          const int cs = p * 16;
          const int cv = 128 + p * 16;
          v8f ga = zero8(), gb = zero8();
          for (int k = 0; k < 64; k += 4) {
            v2f a, b1, b2;
            a.x = Arow[k];
            a.y = Arow[k + 1];
            const float* Bp = Wr2t + (k + hi * 2) * 384 + ln;
            b1.x = Bp[cs]; b1.y = Bp[384 + cs];
            b2.x = Bp[cv]; b2.y = Bp[384 + cv];
            ga = WMMA_F32(a, b1, ga);
            gb = WMMA_F32(a, b2, gb);
          }
          const int f = cs + ln;
#pragma unroll
          for (int j = 0; j < 8; ++j) {
            const int M = et * 16 + j + hi * 8;
            const int s = esi[M], r = eri[M];
            const float T = ey[0][M] * Hd[0][s * 128 + f]
                          + ey[1][M] * Hd[1][s * 128 + f]
                          + ey[2][M] * Hd[2][s * 128 + f];
            lds_atomic_add(&As[r * 128 + f], ga[j] * P1[s * 128 + f] + gb[j] * T);
          }
        } else {      // g_sv (cols 256+vt*16) paired with g_vv (cols 320+vt*16)
          const int vt = p - 8;
          const int cs = 256 + vt * 16;
          const int cv = 320 + vt * 16;
          v8f ga = zero8(), gb = zero8();
          for (int k = 0; k < 64; k += 4) {
            v2f a, b1, b2;
            a.x = Arow[k];
            a.y = Arow[k + 1];
            const float* Bp = Wr2t + (k + hi * 2) * 384 + ln;
            b1.x = Bp[cs]; b1.y = Bp[384 + cs];
            b2.x = Bp[cv]; b2.y = Bp[384 + cv];
            ga = WMMA_F32(a, b1, ga);
            gb = WMMA_F32(a, b2, gb);
          }
          const int v = vt * 16 + ln;
#pragma unroll
          for (int j = 0; j < 8; ++j) {
            const int M = et * 16 + j + hi * 8;
            const int s = esi[M], r = eri[M];
            const float pv = ga[j] * P2[s * 64 + v];
            const float gq = gb[j];
            lds_atomic_add(&Av[0][r * 64 + v], pv * ey[0][M] + gq * Qb[0][s * 64 + v]);
            lds_atomic_add(&Av[1][r * 64 + v], pv * ey[1][M] + gq * Qb[1][s * 64 + v]);
            lds_atomic_add(&Av[2][r * 64 + v], pv * ey[2][M] + gq * Qb[2][s * 64 + v]);
          }
        }
      }
      __syncthreads();
    }

    // ---- scale by 1/N, vv = |A_v|^2 ---------------------------------------
    for (int i = tid; i < 32 * 128; i += 512) As[i] *= (1.f / 32.f);
    for (int i = tid; i < 32 * 64; i += 512) {
      const float a0 = Av[0][i] * (1.f / 32.f);
      const float a1 = Av[1][i] * (1.f / 32.f);
      const float a2 = Av[2][i] * (1.f / 32.f);
      Av[0][i] = a0; Av[1][i] = a1; Av[2][i] = a2;
      P2[i] = a0 * a0 + a1 * a1 + a2 * a2;  // vv (P2 reuse)
    }
    __syncthreads();

    // ---- h_s' = silu([A_s | vv] @ U_s) ------------------------------------
    for (int tile = wave; tile < 16; tile += NW) {
      const int tm = (tile >> 3) << 4;
      const int tn = (tile & 7) << 4;
      v8f acc = zero8();
      for (int k = 0; k < 192; k += 4) {
        const int kk = k + hi * 2;
        v2f a, bb;
        a.x = (kk < 128) ? As[(tm + ln) * 128 + kk] : P2[(tm + ln) * 64 + kk - 128];
        a.y = (kk + 1 < 128) ? As[(tm + ln) * 128 + kk + 1] : P2[(tm + ln) * 64 + kk - 127];
        bb.x = Ust[kk * 128 + tn + ln];
        bb.y = Ust[(kk + 1) * 128 + tn + ln];
        acc = WMMA_F32(a, bb, acc);
      }
#pragma unroll
      for (int j = 0; j < 8; ++j)
        hs[(tm + j + hi * 8) * 128 + tn + ln] = siluf(acc[j]);
    }
    // gate = A_s @ G_sv  (into Hd[0] scratch, 32x64)
    gemm_tiles(As, 128, Gsvt, 64, &Hd[0][0], 64, 32, 64, 128, wave, NW, ln, hi);
    __syncthreads();

    // ---- h_v' = A_v @ U_v + gate * A_v ------------------------------------
    for (int job = wave; job < 24; job += NW) {
      const int c = job / 8;
      const int tile = job % 8;
      const int tm = (tile >> 2) << 4;
      const int tn = (tile & 3) << 4;
      const float* Ac = &Av[c][0];
      v8f acc = zero8();
      for (int k = 0; k < 64; k += 4) {
        v2f a, bb;
        a.x = Ac[(tm + ln) * 64 + k + hi * 2];
        a.y = Ac[(tm + ln) * 64 + k + hi * 2 + 1];
        bb.x = Uvt[(k + hi * 2) * 64 + tn + ln];
        bb.y = Uvt[(k + hi * 2 + 1) * 64 + tn + ln];
        acc = WMMA_F32(a, bb, acc);
      }
#pragma unroll
      for (int j = 0; j < 8; ++j) {
        const int idx = (tm + j + hi * 8) * 64 + tn + ln;
        hv[c][idx] = acc[j] + Hd[0][idx] * Ac[idx];
      }
    }
    __syncthreads();
  }

  // ---- readout -------------------------------------------------------------
  // inv1 = h_s @ W_ro_s  (into As)
  gemm_tiles(hs, 128, W_ro_s, 128, As, 128, 32, 128, 128, wave, NW, ln, hi);
  __syncthreads();

  // out_inv = elu(inv1) @ W_fin + b_fin
  float* oinv = out_inv + b * 32 * 128;
  for (int tile = wave; tile < 16; tile += NW) {
    const int tm = (tile >> 3) << 4;
    const int tn = (tile & 7) << 4;
    v8f acc = zero8();
    for (int k = 0; k < 128; k += 4) {
      const int kk = k + hi * 2;
      v2f a, bb;
      a.x = eluf(As[(tm + ln) * 128 + kk]);
      a.y = eluf(As[(tm + ln) * 128 + kk + 1]);
      bb.x = W_fin[kk * 128 + tn + ln];
      bb.y = W_fin[(kk + 1) * 128 + tn + ln];
      acc = WMMA_F32(a, bb, acc);
    }
#pragma unroll
    for (int j = 0; j < 8; ++j)
      oinv[(tm + j + hi * 8) * 128 + tn + ln] = acc[j] + b_fin[tn + ln];
  }

  // out_vec = h_v @ W_ro_v + com
  float* ovec = out_vec + b * 32 * 64 * 3;
  for (int job = wave; job < 24; job += NW) {
    const int c = job / 8;
    const int tile = job % 8;
    const int tm = (tile >> 2) << 4;
    const int tn = (tile & 3) << 4;
    v8f acc = zero8();
    for (int k = 0; k < 64; k += 4) {
      v2f a, bb;
      a.x = hv[c][(tm + ln) * 64 + k + hi * 2];
      a.y = hv[c][(tm + ln) * 64 + k + hi * 2 + 1];
      bb.x = W_ro_v[(k + hi * 2) * 64 + tn + ln];
      bb.y = W_ro_v[(k + hi * 2 + 1) * 64 + tn + ln];
      acc = WMMA_F32(a, bb, acc);
    }
    const float cm = com[c];
#pragma unroll
    for (int j = 0; j < 8; ++j)
      ovec[((tm + j + hi * 8) * 64 + tn + ln) * 3 + c] = acc[j] + cm;
  }
}

extern "C" void kernel_launch(void* const* d_in, const int* in_sizes, int n_in,
                              void* d_out, int out_size, void* d_ws, size_t ws_size,
                              hipStream_t stream) {
  (void)in_sizes; (void)n_in; (void)d_ws; (void)ws_size; (void)out_size;
  const float* x      = (const float*)d_in[0];
  const float* embed  = (const float*)d_in[1];
  const float* W_r1   = (const float*)d_in[2];
  const float* W_r2   = (const float*)d_in[3];
  const float* Ws1    = (const float*)d_in[4];
  const float* Wvd    = (const float*)d_in[5];
  const float* Ws2    = (const float*)d_in[6];
  const float* Wv1    = (const float*)d_in[7];
  const float* U_s    = (const float*)d_in[8];
  const float* G_sv   = (const float*)d_in[9];
  const float* U_v    = (const float*)d_in[10];
  const float* W_ro_s = (const float*)d_in[11];
  const float* W_ro_v = (const float*)d_in[12];
  const float* W_fin  = (const float*)d_in[13];
  const float* b_fin  = (const float*)d_in[14];
  const int*   snd    = (const int*)d_in[15];
  const int*   rcv    = (const int*)d_in[16];

  float* out_vec = (float*)d_out;                    // (128,32,64,3)
  float* out_inv = out_vec + 128 * 32 * 64 * 3;      // (128,32,128)

  mace_kernel<<<128, 512, 0, stream>>>(x, embed, W_r1, W_r2, Ws1, Wvd, Ws2, Wv1,
                                       U_s, G_sv, U_v, W_ro_s, W_ro_v, W_fin,
                                       b_fin, snd, rcv, out_vec, out_inv);
}